// SEGP_6468220748559
// MI455X (gfx1250) — compile-verified
//
#include <hip/hip_runtime.h>
#include <math.h>

// ---------------- problem constants ----------------
#define NT    320
#define NS    16      // state dim n
#define PIN   8
#define MOUT  8
#define EPSC  1e-3f
#define LTC   0.1f
#define STEPC (1.0f/319.0f)      // TMAX/(NT-1)
#define SQ    6                  // scaling-and-squaring levels
#define SCALE_INV (1.0f/64.0f)   // 2^-SQ

typedef float v2f __attribute__((ext_vector_type(2)));
typedef float v8f __attribute__((ext_vector_type(8)));

// =====================================================================
// Kernel 1: one workgroup. Build A, M = expm(A*step) via scale&square,
// then recur E_k = M E_{k-1}, emitting CE[k]=C E_k, F[k]=CE[k] Sigma0,
// G[k]=CE[k] B into workspace.
// =====================================================================
__global__ void segp_setup(const float* __restrict__ V1_raw,
                           const float* __restrict__ V2_raw,
                           const float* __restrict__ V3_raw,
                           const float* __restrict__ B_w,
                           const float* __restrict__ C_w,
                           const float* __restrict__ covar_x0,
                           float* __restrict__ CEw,   // [NT][8][16]
                           float* __restrict__ Fw,    // [NT][8][16]
                           float* __restrict__ Gw)    // [NT][8][8]
{
    __shared__ float sV1[256], sV1inv[256], sPinv[256], sS[256];
    __shared__ float sA[256], sT1[256], sE[256], sM[256];
    __shared__ float sCov[256], sC[128], sB[128], sCE[128];

    const int t = threadIdx.x;          // 256 threads
    const int r = t >> 4, c = t & 15;

    // V1 = tril(raw,-1) + diag(|diag(raw)|) + eps I ; inverted with +eps I again
    {
        float raw = V1_raw[t];
        sV1[t] = (r > c) ? raw : ((r == c) ? (fabsf(raw) + 2.0f * EPSC) : 0.0f);
    }
    sCov[t] = covar_x0[t];
    if (t < 128) { sC[t] = C_w[t]; sB[t] = B_w[t]; }
    __syncthreads();

    // forward substitution: V1 is lower triangular -> column t of V1^{-1}
    if (t < 16) {
        float x[16];
        for (int i = 0; i < 16; ++i) {
            float s = (i == t) ? 1.0f : 0.0f;
            for (int k = 0; k < i; ++k) s -= sV1[i*16+k] * x[k];
            x[i] = s / sV1[i*16+i];
        }
        for (int i = 0; i < 16; ++i) sV1inv[i*16+t] = x[i];
    }
    __syncthreads();

    // Pinv = V1inv^T V1inv
    {
        float s = 0.0f;
        for (int k = 0; k < 16; ++k) s += sV1inv[k*16+r] * sV1inv[k*16+c];
        sPinv[t] = s;
    }
    // S = -0.5 * tril(V2)tril(V2)^T + (triu(V3,1) - triu(V3,1)^T)
    {
        float s = 0.0f;
        int kk = (r < c) ? r : c;
        for (int k = 0; k <= kk; ++k) s += V2_raw[r*16+k] * V2_raw[c*16+k];
        float v3 = (r < c) ? V3_raw[r*16+c] : ((r > c) ? -V3_raw[c*16+r] : 0.0f);
        sS[t] = -0.5f * s + v3;
    }
    __syncthreads();

    // A*step / 2^SQ
    {
        float s = 0.0f;
        for (int k = 0; k < 16; ++k) s += sPinv[r*16+k] * sS[k*16+c];
        sA[t] = s * (STEPC * SCALE_INV);
    }
    __syncthreads();

    // Taylor series exp(sA) = I + sA + sA^2/2! + ...
    sE[t]  = (r == c) ? 1.0f : 0.0f;
    sT1[t] = sE[t];
    __syncthreads();
    for (int term = 1; term <= 10; ++term) {
        float s = 0.0f;
        for (int k = 0; k < 16; ++k) s += sA[r*16+k] * sT1[k*16+c];
        s /= (float)term;
        __syncthreads();
        sT1[t] = s;
        sE[t] += s;
        __syncthreads();
    }
    // square SQ times -> M = expm(A*step)
    for (int q = 0; q < SQ; ++q) {
        float s = 0.0f;
        for (int k = 0; k < 16; ++k) s += sE[r*16+k] * sE[k*16+c];
        __syncthreads();
        sE[t] = s;
        __syncthreads();
    }
    sM[t] = sE[t];
    sE[t] = (r == c) ? 1.0f : 0.0f;   // E_0 = I
    __syncthreads();

    // recurrence over time steps
    for (int ks = 0; ks < NT; ++ks) {
        if (t < 128) {                 // CE = C @ E   (8x16)
            int m = t >> 4, l = t & 15;
            float s = 0.0f;
            for (int q = 0; q < 16; ++q) s += sC[m*16+q] * sE[q*16+l];
            sCE[t] = s;
            CEw[ks*128 + t] = s;
        }
        __syncthreads();
        if (t < 128) {                 // F = CE @ Sigma0   (8x16)
            int m = t >> 4, l = t & 15;
            float s = 0.0f;
            for (int q = 0; q < 16; ++q) s += sCE[m*16+q] * sCov[q*16+l];
            Fw[ks*128 + t] = s;
        }
        if (t < 64) {                  // G = CE @ B   (8x8)
            int m = t >> 3, p = t & 7;
            float s = 0.0f;
            for (int l = 0; l < 16; ++l) s += sCE[m*16+l] * sB[l*8+p];
            Gw[ks*64 + t] = s;
        }
        // E <- M @ E
        float s = 0.0f;
        for (int k = 0; k < 16; ++k) s += sM[r*16+k] * sE[k*16+c];
        __syncthreads();
        sE[t] = s;
        __syncthreads();
    }
}

// =====================================================================
// Kernel 2: mean_y[i] = C E_i x0 + step*trapz_j w_ij G[i-j] u[j] + D u_d[i]
// one block per i, 32 j-lanes x 8 outputs.
// =====================================================================
__global__ void segp_meany(const float* __restrict__ CEw,
                           const float* __restrict__ Gw,
                           const float* __restrict__ D_w,
                           const float* __restrict__ mean_x0,
                           const float* __restrict__ mean_U,
                           const float* __restrict__ mean_dU,
                           float* __restrict__ outY)
{
    const int i = blockIdx.x;
    const int t = threadIdx.x;               // 256
    __shared__ float red[32][8];
    __shared__ float x0s[16];
    if (t < 16) x0s[t] = mean_x0[t];
    __syncthreads();

    const int m    = t & 7;
    const int lane = t >> 3;                 // 0..31
    float acc = 0.0f;
    for (int j = lane; j <= i; j += 32) {
        float coef = ((j == 0) || (j == i)) ? 0.5f : 1.0f;
        const float* Gk = Gw + (i - j) * 64;
        const float* u  = mean_U + j * 8;
        float s = 0.0f;
        for (int p = 0; p < 8; ++p) s += Gk[m*8+p] * u[p];
        acc += coef * s;
    }
    if (i == 0) acc = 0.0f;                  // w row zero for i==0
    red[lane][m] = acc;
    __syncthreads();

    if (t < 8) {
        float s2 = 0.0f;
        for (int l = 0; l < 32; ++l) s2 += red[l][t];
        float t1 = 0.0f;
        const float* ce = CEw + i*128 + t*16;
        for (int l = 0; l < 16; ++l) t1 += ce[l] * x0s[l];
        float t3 = 0.0f;
        for (int p = 0; p < 8; ++p) t3 += D_w[t*8+p] * mean_dU[i*8+p];
        outY[i*8 + t] = t1 + STEPC * s2 + t3;
    }
}

// =====================================================================
// Kernel 3: covar1[i,j,m,o] = (F[i] @ CE[j]^T)[m,o]  via V_WMMA_F32_16X16X4_F32
// Big-matrix view: rows r=(i*8+m), cols c=(j*8+o); D = Fbig(2560x16) @ CEbig^T.
// One wave per 16x16 tile; K=16 as 4 chained K=4 WMMAs (f32 in, f32 accum).
// =====================================================================
__global__ void segp_covar1(const float* __restrict__ Fw,
                            const float* __restrict__ CEw,
                            float* __restrict__ outC)
{
    const int t      = threadIdx.x;
    const int lane   = t & 31;
    const int tileId = blockIdx.x * 8 + (t >> 5);  // 25600 tiles total
    const int tr = tileId / 160;                   // row tile (covers i = 2tr, 2tr+1)
    const int tc = tileId % 160;                   // col tile (covers j = 2tc, 2tc+1)

    const int half = lane >> 4;                    // K pair select (K∈{0,1} vs {2,3})
    const int sub  = lane & 15;

    // A fragment: A[Mrow,k] = Fbig[tr*16+sub, kb*4 + half*2 + {0,1}]
    const int ia = 2*tr + (sub >> 3);
    const int ma = sub & 7;
    const float* aBase = Fw + ia*128 + ma*16 + half*2;
    // B fragment: B[k,Ncol] = CEbig[tc*16+sub, kb*4 + half*2 + {0,1}]
    const int jb = 2*tc + (sub >> 3);
    const int ob = sub & 7;
    const float* bBase = CEw + jb*128 + ob*16 + half*2;

    v8f d = {};
#pragma unroll
    for (int kb = 0; kb < 4; ++kb) {
        v2f a = *(const v2f*)(aBase + kb*4);   // 8B-aligned (even float offset)
        v2f b = *(const v2f*)(bBase + kb*4);
        d = __builtin_amdgcn_wmma_f32_16x16x4_f32(
                /*neg_a=*/false, a, /*neg_b=*/false, b,
                /*c_mod=*/(short)0, d, /*reuse_a=*/false, /*reuse_b=*/false);
    }

    // D layout: VGPR v: lanes0-15 -> M=v, lanes16-31 -> M=v+8; N = lane&15
    const int n  = lane & 15;
    const int hi = lane >> 4;
    const int j  = 2*tc + (n >> 3);
    const int o  = n & 7;
    const int i0 = 2*tr + hi;
    float* out = outC + (size_t)(i0*320 + j) * 64 + o;   // ((i*N+j)*8+m)*8+o
#pragma unroll
    for (int r2 = 0; r2 < 8; ++r2)
        out[r2*8] = d[r2];                                // m = r2
}

// =====================================================================
// Kernel 4: covar_U[i,j] = exp(-0.5 ((i-j)*step)^2 / LT^2)
// =====================================================================
__global__ void segp_covu(float* __restrict__ outU)
{
    const int idx = blockIdx.x * blockDim.x + threadIdx.x;  // 102400
    const int i = idx / 320, j = idx % 320;
    const float dt = (float)(i - j) * STEPC;
    outU[idx] = __expf(-0.5f * dt * dt * (1.0f / (LTC * LTC)));
}

// =====================================================================
extern "C" void kernel_launch(void* const* d_in, const int* in_sizes, int n_in,
                              void* d_out, int out_size, void* d_ws, size_t ws_size,
                              hipStream_t stream)
{
    (void)in_sizes; (void)n_in; (void)out_size; (void)ws_size;
    const float* V1r  = (const float*)d_in[0];
    const float* V2r  = (const float*)d_in[1];
    const float* V3r  = (const float*)d_in[2];
    const float* B_w  = (const float*)d_in[3];
    const float* C_w  = (const float*)d_in[4];
    const float* D_w  = (const float*)d_in[5];
    const float* mx0  = (const float*)d_in[6];
    const float* cov0 = (const float*)d_in[7];
    const float* mU   = (const float*)d_in[8];
    const float* mdU  = (const float*)d_in[9];

    float* out  = (float*)d_out;
    float* outY = out;                           // 320*8
    float* outC = out + 2560;                    // 320*320*64
    float* outU = out + 2560 + 320*320*64;       // 320*320

    float* ws  = (float*)d_ws;
    float* CEw = ws;                 // NT*128
    float* Fw  = ws + NT*128;        // NT*128
    float* Gw  = ws + 2*NT*128;      // NT*64

    segp_setup <<<1,    256, 0, stream>>>(V1r, V2r, V3r, B_w, C_w, cov0, CEw, Fw, Gw);
    segp_meany <<<NT,   256, 0, stream>>>(CEw, Gw, D_w, mx0, mU, mdU, outY);
    segp_covar1<<<3200, 256, 0, stream>>>(Fw, CEw, outC);   // 25600 tiles / 8 waves per block
    segp_covu  <<<400,  256, 0, stream>>>(outU);
}